// DBToAmplitude_16011638079668
// MI455X (gfx1250) — compile-verified
//
#include <hip/hip_runtime.h>
#include <hip/hip_bf16.h>

// out = 10^x elementwise, fp32 -> fp32 (64*80*20000 = 102.4M elements).
// Memory-bound: ~819 MB traffic -> ~35us floor at 23.3 TB/s; compute (one
// v_exp_f32 + v_mul_f32 per element) hides entirely under memory latency.
//
// Bulk kernel: NO bounds checks (launched only for fully-covered blocks),
// 32-bit indices (buffer < 4 GiB), 4x B128 non-temporal loads/stores per
// thread for memory-level parallelism. Tail kernel: guarded scalar cleanup.

typedef float v4f __attribute__((ext_vector_type(4)));

#ifndef LOG2_10F
#define LOG2_10F 3.3219280948873623f
#endif

__global__ __launch_bounds__(256) void db_to_amp_bulk(
    const v4f* __restrict__ in, v4f* __restrict__ out) {
  constexpr unsigned VPT = 4;    // vec4s per thread
  constexpr unsigned BLOCK = 256;
  const unsigned base = blockIdx.x * (BLOCK * VPT) + threadIdx.x;

  v4f v[VPT];
#pragma unroll
  for (unsigned k = 0; k < VPT; ++k) {
    v[k] = __builtin_nontemporal_load(&in[base + k * BLOCK]);  // b128 TH=NT
  }

#pragma unroll
  for (unsigned k = 0; k < VPT; ++k) {
    v[k].x = __builtin_amdgcn_exp2f(v[k].x * LOG2_10F);
    v[k].y = __builtin_amdgcn_exp2f(v[k].y * LOG2_10F);
    v[k].z = __builtin_amdgcn_exp2f(v[k].z * LOG2_10F);
    v[k].w = __builtin_amdgcn_exp2f(v[k].w * LOG2_10F);
  }

#pragma unroll
  for (unsigned k = 0; k < VPT; ++k) {
    __builtin_nontemporal_store(v[k], &out[base + k * BLOCK]);  // b128 TH=NT
  }
}

// Guarded scalar cleanup for elements [start, n). At most one block-chunk of
// work; never launched for the reference shape (exactly 25,000 full blocks).
__global__ __launch_bounds__(256) void db_to_amp_tail(
    const float* __restrict__ in, float* __restrict__ out, long long start,
    long long n) {
  const long long i = start + threadIdx.x + (long long)blockIdx.x * blockDim.x;
  if (i < n) {
    out[i] = __builtin_amdgcn_exp2f(in[i] * LOG2_10F);
  }
}

extern "C" void kernel_launch(void* const* d_in, const int* in_sizes, int n_in,
                              void* d_out, int out_size, void* d_ws,
                              size_t ws_size, hipStream_t stream) {
  (void)n_in;
  (void)d_ws;
  (void)ws_size;
  (void)in_sizes;

  const float* in = (const float*)d_in[0];
  float* out = (float*)d_out;

  const long long n = (long long)out_size;
  const long long nvec = n >> 2;                    // float4 count
  constexpr long long PER_BLOCK = 256 * 4;          // vec4s per bulk block
  const long long full_blocks = nvec / PER_BLOCK;   // unguarded coverage
  const long long bulk_elems = full_blocks * PER_BLOCK * 4;

  if (full_blocks > 0) {
    db_to_amp_bulk<<<(unsigned)full_blocks, 256, 0, stream>>>(
        (const v4f*)in, (v4f*)out);
  }

  if (bulk_elems < n) {
    const long long rem = n - bulk_elems;
    const long long nblocks = (rem + 255) / 256;
    db_to_amp_tail<<<(unsigned)nblocks, 256, 0, stream>>>(in, out, bulk_elems,
                                                          n);
  }
}